// DiffusionSampler_62861141344409
// MI455X (gfx1250) — compile-verified
//
#include <hip/hip_runtime.h>
#include <math.h>
#include <stdint.h>

// ---------------------------------------------------------------------------
// Diffusion sampler, scan folded analytically:
//   x1[n,j] = Atot*x_T[n,j] + (Cadd*prior[j] + Cg) + Sg*sqrt(glo[j]) * z[n,j]
//   out     = sigmoid( main_xt[0]*x1 + add[0]*prior[j] )   (IEEE inf/NaN kept)
// One N(0,1) draw per element (sum of independent Gaussians == one Gaussian).
// Memory-bound: 80MB in + 80MB out @ 23.3 TB/s ~ 6.9us floor.
// CDNA5 paths: global_load_async_to_lds_b32 + s_wait_asynccnt for LDS table
// staging, global_prefetch_b8 on the input stream, b128 vmem, wave32 blocks.
// ---------------------------------------------------------------------------

#define BETA_1 1e-4
#define BETA_T 0.02

// ---------- Pass 1: fold the T-step scan (fp64, one thread; 100 iterations) ----
__global__ void diff_precompute(const float* __restrict__ graph_var,
                                const float* __restrict__ prior,
                                float* __restrict__ ws, int T) {
    if (threadIdx.x != 0 || blockIdx.x != 0) return;
    float* Pc = ws;          // [T]  Cadd*prior[j] + Cg
    float* Qc = ws + T;      // [T]  Sg*sqrt(glo[j])        (temp: sqrt(glo[j]))
    float* Rc = ws + 2 * T;  // [T]  add[0]*prior[j]        (-inf * prior[j])
    float* Sc = ws + 3 * T;  // scalars: {Atot, m0}

    double ab = 1.0;                       // running inclusive cumprod(alphas)
    double M = 1.0, Cadd = 0.0, Cg = 0.0, V = 0.0;
    double add0 = 0.0, m0 = 0.0;
    for (int t = 0; t < T; ++t) {
        double beta  = BETA_1 + (BETA_T - BETA_1) * (double)t / (double)(T - 1);
        double alpha = 1.0 - beta;
        double sa    = sqrt(alpha);
        double abp   = ab;                 // exclusive cumprod (abp[0] = 1)
        ab *= alpha;                       // inclusive cumprod
        double ka  = (double)t * abp;
        double glo = ka / (ka + 1.0);
        double mt  = (ka + 1.0) / (sa * ka);                    // t=0 -> +inf
        double at  = (sa - 1.0) / sa + (sa - 1.0) / (sa * ka);  // t=0 -> -inf
        Qc[t] = (float)sqrt(glo);          // column noise shape; scaled by Sg below
        if (t == 0) { m0 = mt; add0 = at; continue; }
        // sigma_t_prev = sig[t]*sqrt(1-ab[t]) = sqrt(abp)/(t*abp)*(1-ab)
        double sigprev = sqrt(abp) / (double)t / abp * (1.0 - ab);
        // M == prod_{1<=s<t} main_xt[s] : multiplier applied AFTER step t
        Cadd += M * at;
        Cg   += M * sigprev * (double)graph_var[t];
        V    += M * M * (double)t;         // gvar = (t/T^2)*glo[j]
        M    *= mt;
    }
    double Sg = sqrt(V) / (double)T;
    for (int j = 0; j < T; ++j) {
        Pc[j] = (float)(Cadd * (double)prior[j] + Cg);
        Qc[j] = (float)(Sg * (double)Qc[j]);
        Rc[j] = (float)(add0 * (double)prior[j]);   // -inf*prior (NaN if prior==0)
    }
    Sc[0] = (float)M;    // Atot
    Sc[1] = (float)m0;   // +inf
}

// ---------- deterministic counter-based RNG + Box-Muller -----------------------
__device__ __forceinline__ uint32_t hash32(uint32_t x) {
    x ^= x >> 16; x *= 0x7feb352dU;
    x ^= x >> 15; x *= 0x846ca68bU;
    x ^= x >> 16;
    return x;
}
__device__ __forceinline__ float u01(uint32_t h) {
    return ((float)(h >> 8) + 0.5f) * (1.0f / 16777216.0f);   // (0,1)
}
__device__ __forceinline__ void boxmuller(uint32_t p, float* z0, float* z1) {
    float u1 = u01(hash32(p * 0x9E3779B9u + 0x6A09E667u));
    float u2 = u01(hash32(p * 0x85EBCA6Bu + 0xBB67AE85u));
    float r  = sqrtf(-2.0f * __logf(u1));
    float s, c;
    __sincosf(6.28318530717958647692f * u2, &s, &c);
    *z0 = r * c;
    *z1 = r * s;
}
__device__ __forceinline__ float sigmoidf(float v) {
    return 1.0f / (1.0f + __expf(-v));     // sigmoid(+inf)=1, (-inf)=0, NaN->NaN
}

// ---------- CDNA5 async global->LDS staging (ASYNCcnt path) --------------------
// Per-lane copy: one dword from global address saddr + vaddr to the LDS address in vdst.  The LDS byte
// address is the low 32 bits of the generic pointer (aperture map: addr[31:0]).
__device__ __forceinline__ void async_copy_dword(const float* gbase,
                                                 float* lds_elem,
                                                 uint32_t byte_off) {
    uint32_t lds_addr = (uint32_t)(uintptr_t)lds_elem;
    asm volatile("global_load_async_to_lds_b32 %0, %1, %2"
                 :
                 : "v"(lds_addr), "v"(byte_off),
                   "s"((uint64_t)(uintptr_t)gbase)
                 : "memory");
}
__device__ __forceinline__ void async_wait0() {
    asm volatile("s_wait_asynccnt 0" ::: "memory");
}

// ---------- Pass 2 (vectorized, T % 4 == 0): b128 stream + LDS tables ----------
__global__ void diff_apply_v4(const float* __restrict__ x,
                              const float* __restrict__ ws,
                              float* __restrict__ out,
                              int T, long long nchunks) {
    extern __shared__ float lds[];         // 3*T floats: Pc | Qc | Rc
    const int T3 = 3 * T;
    // Stage coefficient tables via CDNA5 async-to-LDS (per-lane dword scatter).
    for (int i = threadIdx.x; i < T3; i += blockDim.x)
        async_copy_dword(ws, &lds[i], (uint32_t)i * 4u);
    async_wait0();
    __syncthreads();

    const float* Pc = lds;
    const float* Qc = lds + T;
    const float* Rc = lds + 2 * T;
    const float Atot = ws[T3 + 0];
    const float m0   = ws[T3 + 1];
    const int Tq = T >> 2;                 // float4 chunks per row

    const long long stride = (long long)gridDim.x * blockDim.x;
    for (long long c = (long long)blockIdx.x * blockDim.x + threadIdx.x;
         c < nchunks; c += stride) {
        __builtin_prefetch(x + 4 * (c + stride), 0, 1);   // global_prefetch_b8
        float4 xv = ((const float4*)x)[c];                // global_load_b128
        int jq = (int)(c % (long long)Tq);
        float4 p = ((const float4*)Pc)[jq];               // ds_load_b128
        float4 q = ((const float4*)Qc)[jq];
        float4 r = ((const float4*)Rc)[jq];
        float z0, z1, z2, z3;
        uint32_t pp = (uint32_t)c * 2u;
        boxmuller(pp,      &z0, &z1);
        boxmuller(pp + 1u, &z2, &z3);
        float4 o;
        { float x1 = fmaf(Atot, xv.x, fmaf(q.x, z0, p.x)); o.x = sigmoidf(fmaf(m0, x1, r.x)); }
        { float x1 = fmaf(Atot, xv.y, fmaf(q.y, z1, p.y)); o.y = sigmoidf(fmaf(m0, x1, r.y)); }
        { float x1 = fmaf(Atot, xv.z, fmaf(q.z, z2, p.z)); o.z = sigmoidf(fmaf(m0, x1, r.z)); }
        { float x1 = fmaf(Atot, xv.w, fmaf(q.w, z3, p.w)); o.w = sigmoidf(fmaf(m0, x1, r.w)); }
        ((float4*)out)[c] = o;                            // global_store_b128
    }
}

// ---------- Pass 2 (scalar fallback for T % 4 != 0) ----------------------------
__global__ void diff_apply_s(const float* __restrict__ x,
                             const float* __restrict__ ws,
                             float* __restrict__ out,
                             int T, long long total) {
    extern __shared__ float lds[];
    const int T3 = 3 * T;
    for (int i = threadIdx.x; i < T3; i += blockDim.x) lds[i] = ws[i];
    __syncthreads();
    const float* Pc = lds;
    const float* Qc = lds + T;
    const float* Rc = lds + 2 * T;
    const float Atot = ws[T3 + 0];
    const float m0   = ws[T3 + 1];

    const long long stride = (long long)gridDim.x * blockDim.x;
    for (long long e = (long long)blockIdx.x * blockDim.x + threadIdx.x;
         e < total; e += stride) {
        int j = (int)(e % (long long)T);
        float z, zunused;
        boxmuller((uint32_t)e, &z, &zunused);
        float x1 = fmaf(Atot, x[e], fmaf(Qc[j], z, Pc[j]));
        out[e] = sigmoidf(fmaf(m0, x1, Rc[j]));
    }
}

extern "C" void kernel_launch(void* const* d_in, const int* in_sizes, int n_in,
                              void* d_out, int out_size, void* d_ws, size_t ws_size,
                              hipStream_t stream) {
    const float* x     = (const float*)d_in[0];   // [N,T]
    const float* gvar  = (const float*)d_in[1];   // [T]
    const float* prior = (const float*)d_in[2];   // [T]
    float* out = (float*)d_out;
    float* ws  = (float*)d_ws;                    // needs (3*T + 2) floats

    const int T = in_sizes[1];
    const long long total = (long long)in_sizes[0];  // N*T

    diff_precompute<<<1, 32, 0, stream>>>(gvar, prior, ws, T);

    const size_t shbytes = (size_t)(3 * T) * sizeof(float);
    const int threads = 256;                       // 8 wave32s / block
    if ((T & 3) == 0) {
        long long nchunks = total >> 2;
        long long want = (nchunks + threads - 1) / threads;
        int blocks = (int)(want < 1 ? 1 : (want > 4096 ? 4096 : want));
        diff_apply_v4<<<blocks, threads, shbytes, stream>>>(x, ws, out, T, nchunks);
    } else {
        long long want = (total + threads - 1) / threads;
        int blocks = (int)(want < 1 ? 1 : (want > 4096 ? 4096 : want));
        diff_apply_s<<<blocks, threads, shbytes, stream>>>(x, ws, out, T, total);
    }
}